// SGATConv_58875411693627
// MI455X (gfx1250) — compile-verified
//
#include <hip/hip_runtime.h>
#include <hip/hip_bf16.h>
#include <math.h>

// ---------------------------------------------------------------------------
// SGATConv on MI455X (gfx1250, wave32)
//   feat:   (100000, 64) f32      W: (64, 64) f32
//   attn_l/attn_r: (1,4,16) f32   bias: (64,) f32
//   src/dst: (1600000,) i32
//   out:    (100000, 4, 16) f32
// Phase 1: fp32 WMMA GEMM  feat @ W -> feat_src  (V_WMMA_F32_16X16X4_F32)
// Phase 2: head-0 attention scalars el0/er0 (only m[:,0] is used downstream)
// Phase 3: per-edge hard-concrete gate + gather/scatter-add (L2-resident)
// ---------------------------------------------------------------------------

typedef __attribute__((ext_vector_type(2))) float v2f;
typedef __attribute__((ext_vector_type(8))) float v8f;

#define N_NODES   100000
#define IN_FEATS  64
#define OUT_COLS  64          // NUM_HEADS * OUT_FEATS = 4 * 16

// ---- Phase 0: out = bias (broadcast) --------------------------------------
__global__ void sgat_init_kernel(float* __restrict__ out,
                                 const float* __restrict__ bias, int total) {
    int i = blockIdx.x * blockDim.x + threadIdx.x;
    if (i < total) out[i] = bias[i & 63];
}

// ---- Phase 1: feat_src = feat @ W via V_WMMA_F32_16X16X4_F32 --------------
// One wave computes a 16x64 output stripe: 4 head-tiles of 16x16, K swept
// 64/4 = 16 steps.  A-fragment (16x4 f32, 2 VGPRs/lane):
//   lanes 0-15  hold A[M=lane][K=k+0], A[M=lane][K=k+1]
//   lanes 16-31 hold A[M=lane-16][K=k+2], A[M=lane-16][K=k+3]
// B-fragment (4x16 f32, 2 VGPRs/lane), transposed striping:
//   lanes 0-15:  VGPR0=B[k+0][n], VGPR1=B[k+1][n]
//   lanes 16-31: VGPR0=B[k+2][n], VGPR1=B[k+3][n]
// C/D (16x16 f32, 8 VGPRs): VGPR r -> row r (lanes 0-15) / row r+8 (16-31).
__global__ void sgat_gemm_wmma_kernel(const float* __restrict__ feat,
                                      const float* __restrict__ W,
                                      float* __restrict__ feat_src,
                                      int n_nodes) {
    const int wave = (blockIdx.x * blockDim.x + threadIdx.x) >> 5;
    const int lane = threadIdx.x & 31;
    const int row0 = wave * 16;
    if (row0 >= n_nodes) return;

    const int mrow  = lane & 15;   // row within tile (A) / col within tile (B,C)
    const int khalf = lane >> 4;   // 0: K pair {0,1}, 1: K pair {2,3}

    v8f acc0 = {}, acc1 = {}, acc2 = {}, acc3 = {};

    const float* arow = feat + (size_t)(row0 + mrow) * IN_FEATS;

#pragma unroll
    for (int k = 0; k < IN_FEATS; k += 4) {
        const int kk = k + 2 * khalf;
        v2f a;
        a.x = arow[kk + 0];
        a.y = arow[kk + 1];

        const float* bp = W + (size_t)kk * OUT_COLS + mrow;
        v2f b0, b1, b2, b3;
        b0.x = bp[0];           b0.y = bp[OUT_COLS];
        b1.x = bp[16];          b1.y = bp[OUT_COLS + 16];
        b2.x = bp[32];          b2.y = bp[OUT_COLS + 32];
        b3.x = bp[48];          b3.y = bp[OUT_COLS + 48];

        acc0 = __builtin_amdgcn_wmma_f32_16x16x4_f32(false, a, false, b0,
                                                     (short)0, acc0, false, false);
        acc1 = __builtin_amdgcn_wmma_f32_16x16x4_f32(false, a, false, b1,
                                                     (short)0, acc1, false, false);
        acc2 = __builtin_amdgcn_wmma_f32_16x16x4_f32(false, a, false, b2,
                                                     (short)0, acc2, false, false);
        acc3 = __builtin_amdgcn_wmma_f32_16x16x4_f32(false, a, false, b3,
                                                     (short)0, acc3, false, false);
    }

    // Store: lane holds D[r + 8*khalf][tile*16 + mrow] in acc[tile][r]
#pragma unroll
    for (int r = 0; r < 8; ++r) {
        float* op = feat_src + (size_t)(row0 + r + 8 * khalf) * OUT_COLS + mrow;
        op[0]  = acc0[r];
        op[16] = acc1[r];
        op[32] = acc2[r];
        op[48] = acc3[r];
    }
}

// ---- Phase 2: el0/er0 = head-0 attention scalars --------------------------
__global__ void sgat_eler_kernel(const float* __restrict__ feat_src,
                                 const float* __restrict__ attn_l,
                                 const float* __restrict__ attn_r,
                                 float* __restrict__ el0,
                                 float* __restrict__ er0, int n) {
    int i = blockIdx.x * blockDim.x + threadIdx.x;
    if (i >= n) return;
    const float4* row = (const float4*)(feat_src + (size_t)i * OUT_COLS); // head 0
    const float4* al  = (const float4*)attn_l;   // attn_l[0][0][0..15]
    const float4* ar  = (const float4*)attn_r;
    float sl = 0.0f, sr = 0.0f;
#pragma unroll
    for (int j = 0; j < 4; ++j) {
        float4 v = row[j];
        float4 l = al[j];
        float4 r = ar[j];
        sl += v.x * l.x + v.y * l.y + v.z * l.z + v.w * l.w;
        sr += v.x * r.x + v.y * r.y + v.z * r.z + v.w * r.w;
    }
    el0[i] = sl;
    er0[i] = sr;
}

// ---- Phase 3: gated gather + atomic scatter-add ---------------------------
// 16 threads per edge; each handles a float4 slice (coalesced 256B gather).
// Gate (hard-concrete L0): a = clip(sigmoid(x/0.66)*1.2 - 0.1, 0, 1).
__global__ void sgat_edge_kernel(const float* __restrict__ feat_src,
                                 const float* __restrict__ el0,
                                 const float* __restrict__ er0,
                                 const int* __restrict__ src,
                                 const int* __restrict__ dst,
                                 float* __restrict__ out, int n_edges) {
    int t = blockIdx.x * blockDim.x + threadIdx.x;
    int e = t >> 4;
    int j = t & 15;
    if (e >= n_edges) return;

    int s = src[e];
    int d = dst[e];

    float x  = el0[s] + er0[d];
    float sg = 1.0f / (1.0f + __expf(-x * (1.0f / 0.66f)));
    float a  = sg * 1.2f - 0.1f;          // (ZETA-GAMMA)=1.2, GAMMA=-0.1
    a = fminf(fmaxf(a, 0.0f), 1.0f);
    if (a <= 0.0f) return;                // closed gate: contribution exactly 0

    float4 v = ((const float4*)(feat_src + (size_t)s * OUT_COLS))[j];
    float* op = out + (size_t)d * OUT_COLS + 4 * j;
    atomicAdd(op + 0, v.x * a);
    atomicAdd(op + 1, v.y * a);
    atomicAdd(op + 2, v.z * a);
    atomicAdd(op + 3, v.w * a);
}

// ---------------------------------------------------------------------------
extern "C" void kernel_launch(void* const* d_in, const int* in_sizes, int n_in,
                              void* d_out, int out_size, void* d_ws, size_t ws_size,
                              hipStream_t stream) {
    const float* feat   = (const float*)d_in[0];   // (100000, 64)
    const float* W      = (const float*)d_in[1];   // (64, 64)
    const float* attn_l = (const float*)d_in[2];   // (1, 4, 16)
    const float* attn_r = (const float*)d_in[3];   // (1, 4, 16)
    const float* bias   = (const float*)d_in[4];   // (64,)
    const int*   src    = (const int*)d_in[5];     // (1600000,)
    const int*   dst    = (const int*)d_in[6];     // (1600000,)
    float*       out    = (float*)d_out;           // (100000, 4, 16)

    const int n_nodes = in_sizes[0] / IN_FEATS;    // 100000
    const int n_edges = in_sizes[5];               // 1600000
    const int total   = n_nodes * OUT_COLS;        // 6.4M output floats

    // Workspace layout (all offsets 16B aligned):
    //   feat_src : n_nodes*64 f32  (25.6 MB)
    //   el0      : n_nodes   f32
    //   er0      : n_nodes   f32
    char* ws = (char*)d_ws;
    float* feat_src = (float*)ws;
    float* el0      = (float*)(ws + (size_t)n_nodes * OUT_COLS * sizeof(float));
    float* er0      = el0 + n_nodes;

    // Phase 0: out = bias
    {
        int blocks = (total + 255) / 256;
        sgat_init_kernel<<<blocks, 256, 0, stream>>>(out, bias, total);
    }
    // Phase 1: WMMA GEMM (one wave per 16-row tile; 100000 % 16 == 0)
    {
        int waves  = (n_nodes + 15) / 16;          // 6250
        int blocks = (waves + 7) / 8;              // 8 waves / 256-thread block
        sgat_gemm_wmma_kernel<<<blocks, 256, 0, stream>>>(feat, W, feat_src, n_nodes);
    }
    // Phase 2: head-0 attention scalars
    {
        int blocks = (n_nodes + 255) / 256;
        sgat_eler_kernel<<<blocks, 256, 0, stream>>>(feat_src, attn_l, attn_r,
                                                     el0, er0, n_nodes);
    }
    // Phase 3: edge gather / gated scatter-add (L2-resident working set)
    {
        long long threads = (long long)n_edges * 16;
        int blocks = (int)((threads + 255) / 256);
        sgat_edge_kernel<<<blocks, 256, 0, stream>>>(feat_src, el0, er0,
                                                     src, dst, out, n_edges);
    }
}